// EdgePruner_69793218560463
// MI455X (gfx1250) — compile-verified
//
#include <hip/hip_runtime.h>
#include <hip/hip_bf16.h>

// Problem constants
#define B_   16
#define S_   512
#define L_   12
#define H_   12
#define DH_  64
#define D_   768
#define M_   13
#define K_   (L_ * H_ * DH_)   // 9216

// GEMM tiling
#define MT   64
#define NT   64
#define KC   64
#define LDA  66   // padded (even) float stride for A tile -> 8B-aligned b64 lane loads
#define LDB2 80   // packed-B stride in float2 units; 160 floats/row => +32 bank shift
                  // per k-pair row -> the two lane halves use disjoint bank sets

typedef __attribute__((ext_vector_type(2))) float v2f;
typedef __attribute__((ext_vector_type(8))) float v8f;

// ---------------------------------------------------------------------------
// Kernel 1: per-batch constant vector
//   cvec[b][d] = sum_m (1-mlp_mask[b,m]) * modal_mlp[m,d]
//             + sum_p post_bias[p,d]
//             + sum_{p,h} (1-attn_mask[b,p,h]) * modal_attention[p,h,d]
// ---------------------------------------------------------------------------
__global__ __launch_bounds__(256)
void cvec_kernel(const float* __restrict__ mlp_mask,
                 const float* __restrict__ attn_mask,
                 const float* __restrict__ modal_mlp,
                 const float* __restrict__ modal_attention,
                 const float* __restrict__ post_bias,
                 float* __restrict__ cvec) {
    const int b = blockIdx.x;
    for (int d = threadIdx.x; d < D_; d += 256) {
        float acc = 0.0f;
        #pragma unroll
        for (int m = 0; m < M_; ++m)
            acc += (1.0f - mlp_mask[b * M_ + m]) * modal_mlp[m * D_ + d];
        #pragma unroll
        for (int p = 0; p < L_; ++p)
            acc += post_bias[p * D_ + d];
        for (int ph = 0; ph < L_ * H_; ++ph)
            acc += (1.0f - attn_mask[b * (L_ * H_) + ph]) * modal_attention[ph * D_ + d];
        cvec[b * D_ + d] = acc;
    }
}

// ---------------------------------------------------------------------------
// Kernel 2: out[b,s,d] = cvec[b,d] + sum_m mlp_mask[b,m] * mlp_stack[b,s,m,d]
// Pure streaming (327 MB) -> float4 VALU path, one block per (b,s) row.
// ---------------------------------------------------------------------------
__global__ __launch_bounds__(192)
void mlp_base_kernel(const float* __restrict__ mlp_stack,
                     const float* __restrict__ mlp_mask,
                     const float* __restrict__ cvec,
                     float* __restrict__ out) {
    const int r = blockIdx.x;        // b*S + s
    const int b = r >> 9;            // / S_
    const int t = threadIdx.x;       // one float4 of D (768/4 = 192)

    float m[M_];
    #pragma unroll
    for (int i = 0; i < M_; ++i) m[i] = mlp_mask[b * M_ + i];

    const float4* base = (const float4*)(mlp_stack + (size_t)r * M_ * D_);
    float4 acc = ((const float4*)(cvec + (size_t)b * D_))[t];
    #pragma unroll
    for (int i = 0; i < M_; ++i) {
        const float4 v = base[i * (D_ / 4) + t];
        acc.x += m[i] * v.x;
        acc.y += m[i] * v.y;
        acc.z += m[i] * v.z;
        acc.w += m[i] * v.w;
    }
    ((float4*)(out + (size_t)r * D_))[t] = acc;
}

// ---------------------------------------------------------------------------
// Kernel 3: out[r, d] += sum_k (attn_mask-scaled attn_stack)[r,k] * W_O[k,d]
// GEMM [8192 x 9216] x [9216 x 768], fp32 WMMA 16x16x4.
//  - 256 threads (8 wave32), 64x64 output tile, KC=64 chunks (one (p,h) each)
//  - B staged in LDS k-pair-packed: one ds_load_b64 per B fragment, no movs
//  - register-staged prefetch of next chunk overlaps global with WMMA compute
// ---------------------------------------------------------------------------
__global__ __launch_bounds__(256)
void attn_gemm_kernel(const float* __restrict__ attn_stack,
                      const float* __restrict__ attn_mask,
                      const float* __restrict__ W_O,
                      float* __restrict__ out) {
    __shared__ float sA[MT * LDA];                 // 16896 B
    __shared__ float sB[(KC / 2) * LDB2 * 2];      // 20480 B, packed k-pairs

    const int n0   = blockIdx.x * NT;
    const int row0 = blockIdx.y * MT;
    const int b    = row0 >> 9;               // MT=64 divides S_=512: one batch per tile
    const int tid  = threadIdx.x;
    const int w    = tid >> 5;
    const int lane = tid & 31;

    const int lm    = lane & 15;              // 0..15 within half-wave
    const int koff  = (lane >> 4) << 1;       // 0 for lanes 0-15, 2 for lanes 16-31
    const int m0    = (w & 3) * 16;           // wave's 16-row group
    const int cbase = (w >> 2) * 32;          // wave's 2x16-col groups

    // staging coordinates: thread covers rows r0+16j, float4 at col c0
    const int r0 = tid >> 4;                  // 0..15
    const int c0 = (tid & 15) << 2;           // 0..60
    const int ppar = r0 & 1;                  // k parity for packed-B stores

    const float* amask = attn_mask + (size_t)b * (L_ * H_);

    v8f acc0 = {};
    v8f acc1 = {};

    float4 ra[4], rb[4];
    // prologue: fetch chunk 0 into registers
    #pragma unroll
    for (int j = 0; j < 4; ++j) {
        ra[j] = *(const float4*)(attn_stack + (size_t)(row0 + r0 + 16 * j) * K_ + c0);
        rb[j] = *(const float4*)(W_O + (size_t)(r0 + 16 * j) * D_ + n0 + c0);
    }

    for (int kc = 0; kc < K_; kc += KC) {
        const float scale = amask[kc >> 6];   // uniform per chunk -> scalar load

        // stage registers -> LDS (A scaled by attn_mask; B k-pair packed)
        #pragma unroll
        for (int j = 0; j < 4; ++j) {
            float* da = &sA[(r0 + 16 * j) * LDA + c0];
            da[0] = ra[j].x * scale;
            da[1] = ra[j].y * scale;
            da[2] = ra[j].z * scale;
            da[3] = ra[j].w * scale;
            const int p = (r0 + 16 * j) >> 1;
            float* db = &sB[(p * LDB2 + c0) * 2 + ppar];
            db[0] = rb[j].x;
            db[2] = rb[j].y;
            db[4] = rb[j].z;
            db[6] = rb[j].w;
        }
        __syncthreads();

        // prefetch next chunk into registers (overlaps with WMMA compute)
        if (kc + KC < K_) {
            #pragma unroll
            for (int j = 0; j < 4; ++j) {
                ra[j] = *(const float4*)(attn_stack +
                            (size_t)(row0 + r0 + 16 * j) * K_ + kc + KC + c0);
                rb[j] = *(const float4*)(W_O +
                            (size_t)(kc + KC + r0 + 16 * j) * D_ + n0 + c0);
            }
        }

        // 16 WMMA k-steps of 4; A fragment reused across the two N subtiles.
        #pragma unroll
        for (int kk = 0; kk < KC; kk += 4) {
            // A 16x4: lanes 0-15 hold K=kk+0..1, lanes 16-31 hold K=kk+2..3
            const v2f a = *(const v2f*)&sA[(m0 + lm) * LDA + kk + koff];
            // B 4x16: pair row (kk+koff)/2 packed as (B[k][n], B[k+1][n])
            const int p = (kk + koff) >> 1;
            const v2f b0 = *(const v2f*)&sB[(p * LDB2 + cbase + lm) * 2];
            const v2f b1 = *(const v2f*)&sB[(p * LDB2 + cbase + 16 + lm) * 2];
            acc0 = __builtin_amdgcn_wmma_f32_16x16x4_f32(
                false, a, false, b0, (short)0, acc0, false, false);
            acc1 = __builtin_amdgcn_wmma_f32_16x16x4_f32(
                false, a, false, b1, (short)0, acc1, false, false);
        }
        __syncthreads();
    }

    // Epilogue: C/D layout -> VGPR i holds row m0+i (lanes 0-15) / m0+i+8 (16-31)
    const int hi = (lane >> 4) * 8;
    #pragma unroll
    for (int i = 0; i < 8; ++i) {
        const size_t o = (size_t)(row0 + m0 + i + hi) * D_ + n0 + cbase + lm;
        out[o]      += acc0[i];
        out[o + 16] += acc1[i];
    }
}

// ---------------------------------------------------------------------------
extern "C" void kernel_launch(void* const* d_in, const int* in_sizes, int n_in,
                              void* d_out, int out_size, void* d_ws, size_t ws_size,
                              hipStream_t stream) {
    (void)in_sizes; (void)n_in; (void)out_size; (void)ws_size;

    const float* mlp_stack       = (const float*)d_in[0];
    const float* attn_stack      = (const float*)d_in[1];
    const float* mlp_mask        = (const float*)d_in[2];
    const float* attn_mask       = (const float*)d_in[3];
    const float* modal_mlp       = (const float*)d_in[4];
    const float* modal_attention = (const float*)d_in[5];
    const float* W_O             = (const float*)d_in[6];
    const float* post_bias       = (const float*)d_in[7];
    float*       out             = (float*)d_out;
    float*       cvec            = (float*)d_ws;   // B_ * D_ floats = 48 KB

    cvec_kernel<<<B_, 256, 0, stream>>>(mlp_mask, attn_mask, modal_mlp,
                                        modal_attention, post_bias, cvec);

    mlp_base_kernel<<<B_ * S_, 192, 0, stream>>>(mlp_stack, mlp_mask, cvec, out);

    dim3 g3(D_ / NT, (B_ * S_) / MT);   // (12, 128)
    attn_gemm_kernel<<<g3, 256, 0, stream>>>(attn_stack, attn_mask, W_O, out);
}